// MovieSAGE_25555055411666
// MI455X (gfx1250) — compile-verified
//
#include <hip/hip_runtime.h>

typedef float v2f __attribute__((ext_vector_type(2)));
typedef float v8f __attribute__((ext_vector_type(8)));

#define N_NODES 100000
#define N_EDGES 1600000
#define D_IN    128
#define D_HID   128
#define D_OUT   64

// --- degree: one atomicAdd per edge -------------------------------------
__global__ void sage_deg_kernel(const long long* __restrict__ dst,
                                float* __restrict__ deg, int E) {
    int e = blockIdx.x * blockDim.x + threadIdx.x;
    if (e < E) atomicAdd(&deg[(int)dst[e]], 1.0f);
}

// --- scatter-add: one wave32 per edge, one float4 per lane (128 floats) --
__global__ void sage_scatter_kernel(const long long* __restrict__ src,
                                    const long long* __restrict__ dst,
                                    const float* __restrict__ feat,  // [N,128]
                                    float* __restrict__ agg,         // [N,128]
                                    int E) {
    int tid  = blockIdx.x * blockDim.x + threadIdx.x;
    int e    = tid >> 5;
    int lane = tid & 31;
    if (e >= E) return;
    int s = (int)src[e];
    int d = (int)dst[e];
    const float4 v = ((const float4*)(feat + (size_t)s * 128))[lane];
    float* ap = agg + (size_t)d * 128 + lane * 4;
    atomicAdd(ap + 0, v.x);
    atomicAdd(ap + 1, v.y);
    atomicAdd(ap + 2, v.z);
    atomicAdd(ap + 3, v.w);
}

// --- fused SAGE dense part: out = act( (agg/deg)@Wl + x2@Wr + b ) --------
// One wave computes a 16-row x (NT*16)-col tile via V_WMMA_F32_16X16X4_F32.
// K = 128 per pass, two passes (Wl on mean, Wr on x2) into same accumulators.
template <int NT, bool RELU>
__global__ void sage_wmma_gemm_kernel(const float* __restrict__ agg,  // [N,128]
                                      const float* __restrict__ deg,  // [N]
                                      const float* __restrict__ x2,   // [N,128]
                                      const float* __restrict__ Wl,   // [128,NT*16]
                                      const float* __restrict__ Wr,   // [128,NT*16]
                                      const float* __restrict__ bias, // [NT*16]
                                      float* __restrict__ out,        // [N,NT*16]
                                      int n_rows) {
    const int ldb    = NT * 16;
    int waveId = (int)((blockIdx.x * blockDim.x + threadIdx.x) >> 5);
    int lane   = threadIdx.x & 31;
    int half   = lane >> 4;   // 0: K+{0,1}, 1: K+{2,3}  (ISA 16x4 f32 A layout)
    int l15    = lane & 15;
    int m0     = waveId * 16;
    if (m0 >= n_rows) return;            // wave-uniform: EXEC stays all-ones

    int arow = m0 + l15;                 // A-fragment row for this lane
    float dg   = deg[arow];
    float rdeg = 1.0f / fmaxf(dg, 1.0f); // fold mean into the A load

    v8f acc[NT];
#pragma unroll
    for (int t = 0; t < NT; ++t)
        acc[t] = (v8f){0.f, 0.f, 0.f, 0.f, 0.f, 0.f, 0.f, 0.f};

    const float* aptr = agg + (size_t)arow * 128;
    const float* xptr = x2  + (size_t)arow * 128;

    // pass 1: mean @ Wl   (K = 128 -> 32 wmma steps per column tile)
    for (int k = 0; k < 32; ++k) {
        int kk = 4 * k + 2 * half;
        v2f a;
        a.x = aptr[kk]     * rdeg;
        a.y = aptr[kk + 1] * rdeg;
#pragma unroll
        for (int t = 0; t < NT; ++t) {
            int col = t * 16 + l15;
            v2f b;
            b.x = Wl[(size_t)kk       * ldb + col];
            b.y = Wl[(size_t)(kk + 1) * ldb + col];
            acc[t] = __builtin_amdgcn_wmma_f32_16x16x4_f32(
                false, a, false, b, (short)0, acc[t], false, false);
        }
    }
    // pass 2: x2 @ Wr
    for (int k = 0; k < 32; ++k) {
        int kk = 4 * k + 2 * half;
        v2f a;
        a.x = xptr[kk];
        a.y = xptr[kk + 1];
#pragma unroll
        for (int t = 0; t < NT; ++t) {
            int col = t * 16 + l15;
            v2f b;
            b.x = Wr[(size_t)kk       * ldb + col];
            b.y = Wr[(size_t)(kk + 1) * ldb + col];
            acc[t] = __builtin_amdgcn_wmma_f32_16x16x4_f32(
                false, a, false, b, (short)0, acc[t], false, false);
        }
    }
    // store: C/D layout: VGPR r -> M = r (lanes 0-15) / M = r+8 (lanes 16-31)
#pragma unroll
    for (int t = 0; t < NT; ++t) {
        int col  = t * 16 + l15;
        float bb = bias[col];
#pragma unroll
        for (int r = 0; r < 8; ++r) {
            int rowo = m0 + r + 8 * half;
            float v  = acc[t][r] + bb;
            if (RELU) v = fmaxf(v, 0.0f);
            out[(size_t)rowo * ldb + col] = v;
        }
    }
}

extern "C" void kernel_launch(void* const* d_in, const int* in_sizes, int n_in,
                              void* d_out, int out_size, void* d_ws, size_t ws_size,
                              hipStream_t stream) {
    const float*     x   = (const float*)d_in[0];
    const long long* ei  = (const long long*)d_in[1];
    const float*     W1l = (const float*)d_in[2];
    const float*     W1r = (const float*)d_in[3];
    const float*     b1  = (const float*)d_in[4];
    const float*     W2l = (const float*)d_in[5];
    const float*     W2r = (const float*)d_in[6];
    const float*     b2  = (const float*)d_in[7];
    float*           out = (float*)d_out;

    const long long* src = ei;
    const long long* dst = ei + N_EDGES;

    // workspace layout: deg[N] | agg[N*128] | h[N*128]  (~103 MB)
    float* deg = (float*)d_ws;
    float* agg = deg + N_NODES;
    float* h   = agg + (size_t)N_NODES * 128;

    const int scatter_blocks = (N_EDGES * 32 + 255) / 256;  // wave per edge
    const int gemm_waves     = (N_NODES + 15) / 16;         // 6250, exact
    const int gemm_blocks    = (gemm_waves * 32 + 255) / 256;

    // ---- layer 1 ----
    hipMemsetAsync(deg, 0, (size_t)N_NODES * (1 + 128) * sizeof(float), stream);
    sage_deg_kernel<<<(N_EDGES + 255) / 256, 256, 0, stream>>>(dst, deg, N_EDGES);
    sage_scatter_kernel<<<scatter_blocks, 256, 0, stream>>>(src, dst, x, agg, N_EDGES);
    sage_wmma_gemm_kernel<8, true><<<gemm_blocks, 256, 0, stream>>>(
        agg, deg, x, W1l, W1r, b1, h, N_NODES);

    // ---- layer 2 (degree reused) ----
    hipMemsetAsync(agg, 0, (size_t)N_NODES * 128 * sizeof(float), stream);
    sage_scatter_kernel<<<scatter_blocks, 256, 0, stream>>>(src, dst, h, agg, N_EDGES);
    sage_wmma_gemm_kernel<4, false><<<gemm_blocks, 256, 0, stream>>>(
        agg, deg, h, W2l, W2r, b2, out, N_NODES);
}